// kmeans_cluster_38886633898778
// MI455X (gfx1250) — compile-verified
//
#include <hip/hip_runtime.h>
#include <hip/hip_bf16.h>

#define B_N 8192
#define D_N 1024
#define K_N 128
#define LR_C 0.5f
#define EPS_C 1e-8f
#define NEG_MASK_C (-10000.0f)

typedef __attribute__((ext_vector_type(2))) float v2f;
typedef __attribute__((ext_vector_type(4))) float v4f;
typedef __attribute__((ext_vector_type(8))) float v8f;
typedef __attribute__((ext_vector_type(4))) int   v4i;

// ---------------------------------------------------------------------------
// Kernel 1: row norms for datapoints (blocks 0..B-1) and centroids (B..B+K-1)
// norm = max(||row||_2, EPS). Rows are re-read by the WMMA kernel -> keep RT
// (regular temporal) loads so they stay warm in L2.
// ---------------------------------------------------------------------------
__global__ void norms_kernel(const float* __restrict__ X,
                             const float* __restrict__ C,
                             float* __restrict__ xnorm,
                             float* __restrict__ cnorm) {
    __shared__ float red[256];
    const int bid = blockIdx.x;
    const int tid = threadIdx.x;
    const float* row;
    if (bid < B_N) row = X + (size_t)bid * D_N;
    else           row = C + (size_t)(bid - B_N) * D_N;

    // 256 threads * 4 elements = 1024 = D
    const v4f v = ((const v4f*)row)[tid];
    float s = v.x * v.x + v.y * v.y + v.z * v.z + v.w * v.w;
    red[tid] = s;
    __syncthreads();
    for (int off = 128; off > 0; off >>= 1) {
        if (tid < off) red[tid] += red[tid + off];
        __syncthreads();
    }
    if (tid == 0) {
        float n = fmaxf(sqrtf(red[0]), EPS_C);
        if (bid < B_N) xnorm[bid] = n;
        else           cnorm[bid - B_N] = n;
    }
}

// ---------------------------------------------------------------------------
// Kernel 2: WMMA f32 GEMM (16-row slab x all 128 centroids, K=1024 by 4)
// then per-row top-2 of cosine similarity (dot / (xnorm*cnorm)).
// One wave (32 threads) per block; grid = B/16 = 512.
// Emits v_wmma_f32_16x16x4_f32 (exact fp32, matches reference math).
// ---------------------------------------------------------------------------
__global__ void __launch_bounds__(32)
wmma_sim_top2_kernel(const float* __restrict__ X,
                     const float* __restrict__ C,
                     const float* __restrict__ xnorm,
                     const float* __restrict__ cnorm,
                     int* __restrict__ dp_index,
                     int* __restrict__ sec_index,
                     float* __restrict__ dp_max) {
    __shared__ float sim[16 * K_N];

    const int lane  = threadIdx.x;           // 0..31 (wave32)
    const int m0    = blockIdx.x * 16;
    const int halfM = lane & 15;             // row / col within tile
    const int khalf = (lane >> 4) << 1;      // 0 for lanes 0-15, 2 for 16-31

    const float* __restrict__ arow = X + (size_t)(m0 + halfM) * D_N;

    v8f acc[8];
#pragma unroll
    for (int t = 0; t < 8; ++t) acc[t] = (v8f){0.f,0.f,0.f,0.f,0.f,0.f,0.f,0.f};

    for (int k0 = 0; k0 < D_N; k0 += 4) {
        // A-matrix 16x4 f32: lane<16 -> K = k0,k0+1 ; lane>=16 -> K = k0+2,k0+3
        v2f a;
        a.x = arow[k0 + khalf + 0];
        a.y = arow[k0 + khalf + 1];
#pragma unroll
        for (int t = 0; t < 8; ++t) {
            const int col = t * 16 + halfM;  // centroid (N) index
            const float* __restrict__ brow = C + (size_t)col * D_N;
            v2f b;
            b.x = brow[k0 + khalf + 0];
            b.y = brow[k0 + khalf + 1];
            // D = A(16x4) * B(4x16) + C  -- exact f32 accumulate
            acc[t] = __builtin_amdgcn_wmma_f32_16x16x4_f32(
                false, a, false, b, (short)0, acc[t], false, false);
        }
    }

    // C/D layout: VGPR r: lanes 0-15 -> (M=r, N=lane), lanes 16-31 -> (M=r+8, N=lane-16)
    const int mBase = (lane >> 4) * 8;
#pragma unroll
    for (int t = 0; t < 8; ++t) {
#pragma unroll
        for (int r = 0; r < 8; ++r) {
            sim[(mBase + r) * K_N + t * 16 + halfM] = acc[t][r];
        }
    }
    __syncthreads();

    if (lane < 16) {
        const int m = lane;
        const float inv_xn = 1.0f / xnorm[m0 + m];
        float best = -3.4e38f, sec = -3.4e38f;
        int bi = 0, si = 0;
        for (int n = 0; n < K_N; ++n) {
            const float v = sim[m * K_N + n] * inv_xn / cnorm[n];
            if (v > best) { sec = best; si = bi; best = v; bi = n; }
            else if (v > sec) { sec = v; si = n; }
        }
        dp_index[m0 + m]  = bi;
        sec_index[m0 + m] = si;
        dp_max[m0 + m]    = best;
    }
}

// ---------------------------------------------------------------------------
// Kernel 3: dp_centroid / hard_negative gathers. grid = B, 256 threads,
// each thread moves one float4 of the 1024-dim row (256*4 = 1024).
// Centroid table is 512 KB -> stays in L2 (RT loads); outputs are
// write-once 64 MB streams -> non-temporal stores.
// ---------------------------------------------------------------------------
__global__ void gather_kernel(const float* __restrict__ C,
                              const int* __restrict__ dp_index,
                              const int* __restrict__ sec_index,
                              float* __restrict__ dp_centroid,
                              float* __restrict__ hard_negative) {
    const int i   = blockIdx.x;
    const int tid = threadIdx.x;
    const int bi = dp_index[i];
    const int si = sec_index[i];
    const v4f vb = ((const v4f*)(C + (size_t)bi * D_N))[tid];
    const v4f vs = ((const v4f*)(C + (size_t)si * D_N))[tid];
    __builtin_nontemporal_store(vb, (v4f*)(dp_centroid   + (size_t)i * D_N) + tid);
    __builtin_nontemporal_store(vs, (v4f*)(hard_negative + (size_t)i * D_N) + tid);
}

// ---------------------------------------------------------------------------
// Kernel 4: the bandwidth monster. dp_cluster (B,B) and cos_mask (B,B).
// dp_cluster[i][j] = (idx[i]==idx[j] && i!=j)
// cos_mask[i][j]   = dp_cluster[i][j] * (bcs[i][j] > dp_max[i]) * NEG_MASK
// 768 MB of one-touch traffic vs a 192 MB L2 -> NT load for bcs,
// NT stores for both outputs so the reusable working set isn't evicted.
// grid = (B/1024, B), 256 threads, float4 per thread.
// ---------------------------------------------------------------------------
__global__ void mask_kernel(const float* __restrict__ bcs,
                            const int* __restrict__ dp_index,
                            const float* __restrict__ dp_max,
                            float* __restrict__ dp_cluster,
                            float* __restrict__ cos_mask) {
    const int i  = blockIdx.y;
    const int j0 = (blockIdx.x * 256 + threadIdx.x) * 4;

    const int   myidx = dp_index[i];
    const float mymax = dp_max[i];

    const v4i jidx = *(const v4i*)(dp_index + j0);
    const v4f bv   = __builtin_nontemporal_load(
                         (const v4f*)(bcs + (size_t)i * B_N + j0));

    v4f dc, cm;
    {
        const bool s = (jidx.x == myidx) && (j0 + 0 != i);
        dc.x = s ? 1.0f : 0.0f;
        cm.x = (s && (bv.x > mymax)) ? NEG_MASK_C : 0.0f;
    }
    {
        const bool s = (jidx.y == myidx) && (j0 + 1 != i);
        dc.y = s ? 1.0f : 0.0f;
        cm.y = (s && (bv.y > mymax)) ? NEG_MASK_C : 0.0f;
    }
    {
        const bool s = (jidx.z == myidx) && (j0 + 2 != i);
        dc.z = s ? 1.0f : 0.0f;
        cm.z = (s && (bv.z > mymax)) ? NEG_MASK_C : 0.0f;
    }
    {
        const bool s = (jidx.w == myidx) && (j0 + 3 != i);
        dc.w = s ? 1.0f : 0.0f;
        cm.w = (s && (bv.w > mymax)) ? NEG_MASK_C : 0.0f;
    }
    __builtin_nontemporal_store(dc, (v4f*)(dp_cluster + (size_t)i * B_N + j0));
    __builtin_nontemporal_store(cm, (v4f*)(cos_mask   + (size_t)i * B_N + j0));
}

// ---------------------------------------------------------------------------
// Kernel 5: per-cluster reduction + centroid update. grid = K, 256 threads,
// each thread owns 4 dims. dp_index staged through LDS in chunks of 256 so
// the index scan is a broadcast LDS read, not 8192 global loads per thread.
// Deterministic accumulation order (i ascending).
// ---------------------------------------------------------------------------
__global__ void centroid_update_kernel(const float* __restrict__ X,
                                       const float* __restrict__ C,
                                       const int* __restrict__ dp_index,
                                       float* __restrict__ new_centroid) {
    __shared__ int sidx[256];
    const int k   = blockIdx.x;
    const int tid = threadIdx.x;

    v4f acc = (v4f){0.f, 0.f, 0.f, 0.f};
    int count = 0;

    for (int base = 0; base < B_N; base += 256) {
        __syncthreads();
        sidx[tid] = dp_index[base + tid];
        __syncthreads();
        for (int t = 0; t < 256; ++t) {
            if (sidx[t] == k) {
                const v4f v = ((const v4f*)(X + (size_t)(base + t) * D_N))[tid];
                acc.x += v.x; acc.y += v.y; acc.z += v.z; acc.w += v.w;
                ++count;
            }
        }
    }

    const v4f c = ((const v4f*)(C + (size_t)k * D_N))[tid];
    const float notup = (count == 0) ? 1.0f : 0.0f;
    v4f o;
    o.x = LR_C * (acc.x + c.x * notup) + (1.0f - LR_C) * c.x;
    o.y = LR_C * (acc.y + c.y * notup) + (1.0f - LR_C) * c.y;
    o.z = LR_C * (acc.z + c.z * notup) + (1.0f - LR_C) * c.z;
    o.w = LR_C * (acc.w + c.w * notup) + (1.0f - LR_C) * c.w;
    ((v4f*)(new_centroid + (size_t)k * D_N))[tid] = o;
}

// ---------------------------------------------------------------------------
extern "C" void kernel_launch(void* const* d_in, const int* in_sizes, int n_in,
                              void* d_out, int out_size, void* d_ws, size_t ws_size,
                              hipStream_t stream) {
    (void)in_sizes; (void)n_in; (void)out_size; (void)ws_size;

    const float* X   = (const float*)d_in[0];   // (B, D)
    const float* C   = (const float*)d_in[1];   // (K, D)
    const float* BCS = (const float*)d_in[2];   // (B, B)

    // Output tuple, concatenated flat in return order.
    float* out = (float*)d_out;
    float* dp_centroid   = out;                                         // B*D
    float* dp_cluster    = dp_centroid + (size_t)B_N * D_N;             // B*B
    float* hard_negative = dp_cluster + (size_t)B_N * B_N;              // B*D
    float* cos_mask      = hard_negative + (size_t)B_N * D_N;           // B*B
    float* new_centroid  = cos_mask + (size_t)B_N * B_N;                // K*D

    // Workspace layout.
    char* ws = (char*)d_ws;
    float* xnorm   = (float*)ws;                     ws += (size_t)B_N * sizeof(float);
    float* cnorm   = (float*)ws;                     ws += (size_t)K_N * sizeof(float);
    int*   dpidx   = (int*)ws;                       ws += (size_t)B_N * sizeof(int);
    int*   secidx  = (int*)ws;                       ws += (size_t)B_N * sizeof(int);
    float* dpmax   = (float*)ws;

    norms_kernel<<<B_N + K_N, 256, 0, stream>>>(X, C, xnorm, cnorm);

    wmma_sim_top2_kernel<<<B_N / 16, 32, 0, stream>>>(X, C, xnorm, cnorm,
                                                      dpidx, secidx, dpmax);

    gather_kernel<<<B_N, 256, 0, stream>>>(C, dpidx, secidx,
                                           dp_centroid, hard_negative);

    dim3 mgrid(B_N / 1024, B_N);
    mask_kernel<<<mgrid, 256, 0, stream>>>(BCS, dpidx, dpmax,
                                           dp_cluster, cos_mask);

    centroid_update_kernel<<<K_N, 256, 0, stream>>>(X, C, dpidx, new_centroid);
}